// PrometheusV4Enhanced_8358006358357
// MI455X (gfx1250) — compile-verified
//
#include <hip/hip_runtime.h>
#include <stdint.h>

// ---------------------------------------------------------------------------
// out[b,h,w,c] = x[b,h,w,c] + (c < 256 ? spatial_pe[h,w,c]
//                                      : pattern_pe[idx[b,h,w], c-256])
//
// Bandwidth-bound (0.125 FLOP/byte): ~945 MB over 23.3 TB/s => ~41 us floor.
// Block-level split: half the grid does the spatial half (global b128 only),
// half does the pattern half (TDM-staged 64 KB LDS table + ds_load_b128),
// so the compiler cannot if-convert the gather into a flat load.
// ---------------------------------------------------------------------------

typedef float        v4f __attribute__((ext_vector_type(4)));
typedef unsigned int v4u __attribute__((ext_vector_type(4)));
typedef int          v4i __attribute__((ext_vector_type(4)));
typedef int          v8i __attribute__((ext_vector_type(8)));

#define NUM_PATTERNS 64
#define HALF_D       256                        // d_model/2 floats
#define PAT_FLOATS   (NUM_PATTERNS * HALF_D)    // 16384 floats = 64 KB
#define PAT_VEC      (PAT_FLOATS / 4)           // 4096 float4s
#define PIX_HW       900                        // 30*30 pixels per batch
#define HQPIX        64                         // float4s per half-pixel (256/4)

__global__ __launch_bounds__(256)
void pe_add_kernel(const float* __restrict__ x,
                   const int*   __restrict__ pidx,
                   const float* __restrict__ spatial,
                   const float* __restrict__ pattern,
                   float*       __restrict__ out,
                   unsigned int halfQ)          // float4s per half = totalQ/2
{
    __shared__ v4f pat_lds[PAT_VEC];            // 64 KB of the 320 KB WGP LDS

    const unsigned tid  = threadIdx.x;
    const unsigned G    = gridDim.x;
    const unsigned half = G >> 1;

    const v4f* __restrict__ xv = (const v4f*)x;
    v4f*       __restrict__ ov = (v4f*)out;

    if (blockIdx.x < half) {
        // ------------- Spatial half: c in [0,256) -> float4s c4 in [0,64) --
        const v4f* __restrict__ sv = (const v4f*)spatial;
        const unsigned stride = half * blockDim.x;
        // stride is a multiple of 64, so c4 is invariant under s += stride:
        // the next iteration's q is exactly q + 2*stride (in float4s).
        const unsigned qstep = stride << 1;
        for (unsigned s = blockIdx.x * blockDim.x + tid; s < halfQ; s += stride) {
            const unsigned pix = s >> 6;
            const unsigned c4  = s & (HQPIX - 1u);
            const unsigned q   = (pix << 7) | c4;            // global float4 idx
            __builtin_prefetch(&xv[q + qstep], 0, 0);        // next-iter x line
            const v4f xval = xv[q];                          // global_load_b128
            const v4f pe   = sv[(pix % PIX_HW) * 64u + c4];  // L2-resident 0.9 MB
            ov[q] = xval + pe;                               // global_store_b128
        }
    } else {
        // ------------- Pattern half: c in [256,512) -----------------------
#if __has_builtin(__builtin_amdgcn_tensor_load_to_lds)
        // Stage pattern_pe (64 KB contiguous) into LDS with the TDM.
        // D# per CDNA5 ISA ch.8: 1D tile, data_size=4B, tile_dim0=16384.
        // TDM ignores EXEC and issues once per wave -> only wave 0 issues.
        if (tid < 32u) {
            const uint64_t ga      = (uint64_t)(uintptr_t)pattern;
            const unsigned lds_off = (unsigned)(uintptr_t)(&pat_lds[0]);

            v4u g0;
            g0.x = 1u;                                   // count=1, no gather
            g0.y = lds_off;                              // lds_addr (bytes)
            g0.z = (unsigned)(ga & 0xFFFFFFFFu);         // global_addr[31:0]
            g0.w = (unsigned)((ga >> 32) & 0x01FFFFFFu)  // global_addr[56:32]
                 | (2u << 30);                           // type = 2 ("image")

            v8i g1;
            g1[0] = (int)0x00020000u;                    // data_size=2 (4 B)
            g1[1] = (int)((PAT_FLOATS & 0xFFFF) << 16);  // tensor_dim0[15:0]
            g1[2] = (int)((PAT_FLOATS >> 16) | (1u << 16)); // dim0 hi | dim1=1
            g1[3] = (int)((PAT_FLOATS & 0xFFFF) << 16);  // tile_dim0 = 16384
            g1[4] = (int)1u;                             // tile_dim1=1, tile_dim2=0
            g1[5] = (int)PAT_FLOATS;                     // tensor_dim0_stride
            g1[6] = 0;
            g1[7] = 0;

            v4i gz = (v4i)0;                             // groups 2/3 unused
#if defined(__clang_major__) && (__clang_major__ >= 23)
            v8i gzz = (v8i)0;
            __builtin_amdgcn_tensor_load_to_lds(g0, g1, gz, gz, gzz, 0);
#else
            __builtin_amdgcn_tensor_load_to_lds(g0, g1, gz, gz, 0);
#endif
            __builtin_amdgcn_s_wait_tensorcnt(0);        // TENSORcnt==0
        }
        __syncthreads();                                 // publish LDS
#else
        for (unsigned i = tid; i < PAT_VEC; i += blockDim.x)
            pat_lds[i] = ((const v4f*)pattern)[i];
        __syncthreads();
#endif

        const unsigned stride = (G - half) * blockDim.x;
        const unsigned qstep  = stride << 1;             // next-iter q delta
        for (unsigned s = (blockIdx.x - half) * blockDim.x + tid; s < halfQ; s += stride) {
            const unsigned pix = s >> 6;
            const unsigned c4  = s & (HQPIX - 1u);
            const unsigned q   = (pix << 7) | (64u + c4);    // second half of pixel
            __builtin_prefetch(&xv[q + qstep], 0, 0);        // next-iter x line
            const v4f xval = xv[q];                          // global_load_b128
            // All 32 lanes of a wave share pix -> one uniform idx per wave.
            const unsigned id = ((unsigned)pidx[pix]) & (NUM_PATTERNS - 1u);
            const v4f pe = pat_lds[id * 64u + c4];           // ds_load_b128
            ov[q] = xval + pe;                               // global_store_b128
        }
    }
}

extern "C" void kernel_launch(void* const* d_in, const int* in_sizes, int n_in,
                              void* d_out, int out_size, void* d_ws, size_t ws_size,
                              hipStream_t stream) {
    const float* x       = (const float*)d_in[0];
    const int*   pidx    = (const int*)  d_in[1];   // pattern_indices
    const float* spatial = (const float*)d_in[2];   // [30,30,256]
    const float* pattern = (const float*)d_in[3];   // [64,256]
    float*       out     = (float*)d_out;

    const unsigned halfQ = (unsigned)(out_size / 8);    // 14,745,600 float4s per half

    // Persistent grid: 2048 blocks x 256 threads (8 wave32s). Blocks [0,1024)
    // sweep the spatial half; blocks [1024,2048) TDM-stage the 64 KB pattern
    // table once (~56 iterations amortize it) and sweep the pattern half.
    dim3 grid(2048), block(256);
    pe_add_kernel<<<grid, block, 0, stream>>>(x, pidx, spatial, pattern, out, halfQ);
}